// UNet_6708738916786
// MI455X (gfx1250) — compile-verified
//
#include <hip/hip_runtime.h>
#include <hip/hip_bf16.h>

// Problem constants (from reference)
#define NV_V  1000000
#define NE_E  3000000
#define NDEG  6
#define TILES_PER_WAVE 4

typedef __attribute__((ext_vector_type(2))) float v2f;
typedef __attribute__((ext_vector_type(4))) float v4f;
typedef __attribute__((ext_vector_type(8))) float v8f;
typedef __attribute__((ext_vector_type(2))) int   v2i;

// ---------------------------------------------------------------------------
// Kernel 1: tiny time-MLP -> per-output-channel shift
//   s[c]      = (silu(te@W1+b1) @ W2 + b2) @ Wte + bte
//   shift2[o] = sum_c Wc[o,c]*s[c] + bc[o]
// One block, 64 threads (TDIM = 64).
// ---------------------------------------------------------------------------
__global__ void __launch_bounds__(64) time_mlp_kernel(
    const float* __restrict__ t,
    const float* __restrict__ W1,  const float* __restrict__ b1,
    const float* __restrict__ W2,  const float* __restrict__ b2,
    const float* __restrict__ Wte, const float* __restrict__ bte,
    const float* __restrict__ Wc,  const float* __restrict__ bc,
    float* __restrict__ shift2) {
    __shared__ float te[16];
    __shared__ float h[64];
    __shared__ float t2[64];
    __shared__ float s[6];
    const int j = threadIdx.x;

    if (j < 8) {
        const float freq = expf((float)j * (-logf(10000.0f) / 7.0f));
        const float emb  = t[0] * freq;
        te[j]     = sinf(emb);
        te[j + 8] = cosf(emb);
    }
    __syncthreads();
    {   // h = silu(te @ W1 + b1), W1 is (16,64) row-major
        float acc = b1[j];
        #pragma unroll
        for (int i = 0; i < 16; ++i) acc += te[i] * W1[i * 64 + j];
        h[j] = acc / (1.0f + expf(-acc));
    }
    __syncthreads();
    {   // t2 = h @ W2 + b2, W2 is (64,64)
        float acc = b2[j];
        for (int k = 0; k < 64; ++k) acc += h[k] * W2[k * 64 + j];
        t2[j] = acc;
    }
    __syncthreads();
    if (j < 6) {  // s = t2 @ Wte + bte, Wte is (64,6)
        float acc = bte[j];
        for (int k = 0; k < 64; ++k) acc += t2[k] * Wte[k * 6 + j];
        s[j] = acc;
    }
    __syncthreads();
    if (j < 6) {  // fold channel shift through the 6x6 mix
        float acc = bc[j];
        #pragma unroll
        for (int c = 0; c < 6; ++c) acc += Wc[j * 6 + c] * s[c];
        shift2[j] = acc;
    }
}

// ---------------------------------------------------------------------------
// Kernel 2: channel mix via V_WMMA_F32_16X16X4_F32.
//   feT[e][slot(o)] = sum_c Wc[o,c]*x[c,e] + shift2[o]
// D(16x16) = A(16x4) x B(4x16) + C, chained twice to cover K=6 (zero-padded
// to 8). M = e (16 consecutive columns), N = o (output channel).
// Each wave processes TILES_PER_WAVE consecutive 16-column tiles; the B
// operand (Wc^T) and shift broadcast are loop-invariant and hoisted.
// All zero-padding is done with branch-free selects (in-bounds loads for
// every lane) so EXEC stays all-ones and the loads clause cleanly.
//
// feT is padded to 8 floats/column: {c0,c1,c2,_,c3,c4,c5,_} so the gather
// kernel reads one aligned b128 per edge endpoint.
//
// f32 operand layouts (ISA 7.12.2):
//   A 16x4 : lane m(=lane&15); VGPR0 holds K=0 (lanes<16) / K=2 (lanes>=16),
//            VGPR1 holds K=1 / K=3.  B mirrors with N on lanes.
//   D      : VGPR j -> M = j + 8*(lane>=16), N = lane&15.
// ---------------------------------------------------------------------------
__global__ void __launch_bounds__(256) chanmix_wmma_kernel(
    const float* __restrict__ x, const float* __restrict__ Wc,
    const float* __restrict__ shift2, float* __restrict__ feT, int ntiles) {
    const int lane   = threadIdx.x & 31;
    const int waveId = blockIdx.x * (blockDim.x >> 5) + (threadIdx.x >> 5);
    const int tile0  = waveId * TILES_PER_WAVE;
    if (tile0 >= ntiles) return;          // wave-uniform: EXEC stays all-ones

    const int hi = lane >> 4;             // half-wave select
    const int m  = lane & 15;
    const int o  = m;                     // D column = output channel

    // --- loop-invariant B operands: B[c][o] = Wc[o][c] (clamped o reads) ---
    const int oc = (o < 6) ? o : 0;
    v2f b0, b1;
    b0.x = Wc[oc * 6 + (0 + 2 * hi)];     // K=0 / K=2
    b0.y = Wc[oc * 6 + (1 + 2 * hi)];     // K=1 / K=3
    const float wc4 = Wc[oc * 6 + 4];
    const float wc5 = Wc[oc * 6 + 5];
    b1.x = hi ? 0.0f : wc4;               // K=4 / K=6(pad) -> cndmask
    b1.y = hi ? 0.0f : wc5;               // K=5 / K=7(pad) -> cndmask

    // --- loop-invariant C seed: C[m][o] = shift2[o] ---
    const float sh = (o < 6) ? shift2[o] : 0.0f;

    const int  slot   = o + (o >= 3 ? 1 : 0);   // {0,1,2,_,3,4,5,_}
    const bool active = (o < 6);

    #pragma unroll
    for (int tt = 0; tt < TILES_PER_WAVE; ++tt) {
        const int tile = tile0 + tt;
        if (tile >= ntiles) break;        // uniform (ntiles % TPW == 0 anyway)
        const long eBase = (long)tile * 16;
        const long e     = eBase + m;

        // A operands: x columns (stream-once -> non-temporal, all in-bounds)
        v2f a0, a1;
        a0.x = __builtin_nontemporal_load(x + (long)(0 + 2 * hi) * NE_E + e);
        a0.y = __builtin_nontemporal_load(x + (long)(1 + 2 * hi) * NE_E + e);
        const float x4 = __builtin_nontemporal_load(x + (long)4 * NE_E + e);
        const float x5 = __builtin_nontemporal_load(x + (long)5 * NE_E + e);
        a1.x = hi ? 0.0f : x4;            // K=4 / K=6(pad) -> cndmask
        a1.y = hi ? 0.0f : x5;            // K=5 / K=7(pad) -> cndmask

        v8f c = {sh, sh, sh, sh, sh, sh, sh, sh};
        c = __builtin_amdgcn_wmma_f32_16x16x4_f32(false, a0, false, b0,
                                                  (short)0, c, false, false);
        c = __builtin_amdgcn_wmma_f32_16x16x4_f32(false, a1, false, b1,
                                                  (short)0, c, false, false);

        if (active) {
            float* dst = feT + (eBase + 8 * hi) * 8 + slot;
            #pragma unroll
            for (int j = 0; j < 8; ++j)   // D: M = j + 8*hi
                dst[j * 8] = c[j];
        }
    }
}

// ---------------------------------------------------------------------------
// Kernel 3: segment-mean gather. One thread per vertex.
//   out[v, c] = (1/nvs[v]) * sum_{j<6} feT[ve_in[6v+j]][4*vei[6v+j] + c]
// Index loads are NT (streamed once); the 6 random b128 gathers hit the
// 96 MB feT buffer resident in the 192 MB L2; outputs stored NT.
// ---------------------------------------------------------------------------
__global__ void __launch_bounds__(256) gather_mean_kernel(
    const int* __restrict__ vei, const int* __restrict__ ve_in,
    const float* __restrict__ nvs, const float* __restrict__ feT,
    float* __restrict__ out) {
    const int v = blockIdx.x * blockDim.x + threadIdx.x;
    if (v >= NV_V) return;

    const v2i* pg = (const v2i*)(vei   + 6l * v);
    const v2i* pe = (const v2i*)(ve_in + 6l * v);
    const v2i g01 = __builtin_nontemporal_load(pg);
    const v2i g23 = __builtin_nontemporal_load(pg + 1);
    const v2i g45 = __builtin_nontemporal_load(pg + 2);
    const v2i e01 = __builtin_nontemporal_load(pe);
    const v2i e23 = __builtin_nontemporal_load(pe + 1);
    const v2i e45 = __builtin_nontemporal_load(pe + 2);

    float s0 = 0.0f, s1 = 0.0f, s2 = 0.0f;
    const int g[6] = {g01.x, g01.y, g23.x, g23.y, g45.x, g45.y};
    const int e[6] = {e01.x, e01.y, e23.x, e23.y, e45.x, e45.y};
    #pragma unroll
    for (int j = 0; j < 6; ++j) {
        const v4f q = *(const v4f*)(feT + (long)e[j] * 8 + 4 * g[j]);
        s0 += q.x; s1 += q.y; s2 += q.z;
    }

    const float inv = 1.0f / nvs[v];
    const long  o   = 3l * v;
    __builtin_nontemporal_store(s0 * inv, out + o + 0);
    __builtin_nontemporal_store(s1 * inv, out + o + 1);
    __builtin_nontemporal_store(s2 * inv, out + o + 2);
}

// ---------------------------------------------------------------------------
extern "C" void kernel_launch(void* const* d_in, const int* in_sizes, int n_in,
                              void* d_out, int out_size, void* d_ws, size_t ws_size,
                              hipStream_t stream) {
    const float* x   = (const float*)d_in[0];
    const float* t   = (const float*)d_in[1];
    const float* W1  = (const float*)d_in[2];
    const float* b1  = (const float*)d_in[3];
    const float* W2  = (const float*)d_in[4];
    const float* b2  = (const float*)d_in[5];
    const float* Wte = (const float*)d_in[6];
    const float* bte = (const float*)d_in[7];
    const float* Wc  = (const float*)d_in[8];
    const float* bc  = (const float*)d_in[9];
    const float* nvs = (const float*)d_in[10];
    const int*   vei   = (const int*)d_in[11];
    const int*   ve_in = (const int*)d_in[12];
    // d_in[13] (nvsi) is implied by the fixed-degree layout; unused.

    float* shift2 = (float*)d_ws;                         // 6 floats
    float* feT    = (float*)((char*)d_ws + 256);          // 8*E floats = 96 MB

    time_mlp_kernel<<<1, 64, 0, stream>>>(t, W1, b1, W2, b2, Wte, bte, Wc, bc,
                                          shift2);

    const int ntiles  = NE_E / 16;                        // 187500 (exact)
    const int nwaves  = (ntiles + TILES_PER_WAVE - 1) / TILES_PER_WAVE;
    const int wavesPB = 256 / 32;
    const int blocks2 = (nwaves + wavesPB - 1) / wavesPB;
    chanmix_wmma_kernel<<<blocks2, 256, 0, stream>>>(x, Wc, shift2, feT, ntiles);

    const int blocks3 = (NV_V + 255) / 256;
    gather_mean_kernel<<<blocks3, 256, 0, stream>>>(vei, ve_in, nvs, feT,
                                                    (float*)d_out);
}